// RHF_35708358099217
// MI455X (gfx1250) — compile-verified
//
#include <hip/hip_runtime.h>

#define NBF    96
#define NBF2   (NBF * NBF)     // 9216
#define NDOCC  24
#define N_ITERS 20
#define NSWEEPS 8

typedef __attribute__((ext_vector_type(2))) float v2f;
typedef __attribute__((ext_vector_type(4))) float v4f;
typedef __attribute__((ext_vector_type(8))) float v8f;

__device__ inline float dot4(v4f a, v4f b) {
  return a.x * b.x + a.y * b.y + a.z * b.z + a.w * b.w;
}

// --------------------------------------------------------------------------
// JK kernel: fused J/K contraction, one streaming pass over G (340 MB).
// grid (96, 4), block 768 (24 waves). Block (a, slice): b = slice*24 + wave.
//   J[a,b] = sum_{c,d} G[a,b,c,d] * D[c,d]
//   K[a,c] = sum_{b,d} G[a,b,c,d] * D[b,d]
// G is consumed in b128 quanta: a 4-row chunk (4x96 floats) is 96 float4s;
// lane l owns quads l, l+32, l+64. Quad's d-position (t%24) is lane-invariant
// across chunks, so the D[b] quads hoist out of the loop. K partials use a
// per-lane-banked LDS accumulator via ds_add_f32 (no bank aliasing).
// --------------------------------------------------------------------------
__global__ __launch_bounds__(768) void jk_kernel(const float* __restrict__ G,
                                                 const float* __restrict__ D,
                                                 float* __restrict__ Jg,
                                                 float* __restrict__ Kg) {
  __shared__ v4f   Dl4[NBF2 / 4];   // 36 KB: D resident (quad layout)
  __shared__ float Kacc[NBF * 32];  // 12 KB: K partials [c][lane]
  __shared__ float Jl[24];
  const int tid = threadIdx.x;
  const int a = blockIdx.x, slice = blockIdx.y;

  const v4f* D4 = (const v4f*)D;
  for (int i = tid; i < NBF2 / 4; i += 768) Dl4[i] = D4[i];
  for (int i = tid; i < NBF * 32; i += 768) Kacc[i] = 0.0f;
  __syncthreads();

  const int wave = tid >> 5, lane = tid & 31;
  const int b = slice * 24 + wave;
  const v4f* Gab4 = (const v4f*)(G + (size_t)a * NBF2 * NBF + (size_t)b * NBF2);

  // lane-invariant d-quad positions and c-row offsets within a chunk
  const int m0 = lane % 24, m1 = (lane + 8) % 24, m2 = (lane + 16) % 24;
  const int r0 = lane / 24, r1 = (lane + 32) / 24, r2 = (lane + 64) / 24;

  const v4f db0 = Dl4[b * 24 + m0];
  const v4f db1 = Dl4[b * 24 + m1];
  const v4f db2 = Dl4[b * 24 + m2];
  float jacc = 0.0f;

#pragma unroll 2
  for (int blk = 0; blk < 24; ++blk) {  // 24 chunks of 4 c-rows
    const int base = blk * 96;
    v4f g0 = __builtin_nontemporal_load(&Gab4[base + lane]);
    v4f g1 = __builtin_nontemporal_load(&Gab4[base + lane + 32]);
    v4f g2 = __builtin_nontemporal_load(&Gab4[base + lane + 64]);
    const int c0 = blk * 4 + r0, c1 = blk * 4 + r1, c2 = blk * 4 + r2;
    v4f dc0 = Dl4[c0 * 24 + m0];
    v4f dc1 = Dl4[c1 * 24 + m1];
    v4f dc2 = Dl4[c2 * 24 + m2];
    jacc += dot4(g0, dc0) + dot4(g1, dc1) + dot4(g2, dc2);
    atomicAdd(&Kacc[c0 * 32 + lane], dot4(g0, db0));  // ds_add_f32
    atomicAdd(&Kacc[c1 * 32 + lane], dot4(g1, db1));
    atomicAdd(&Kacc[c2 * 32 + lane], dot4(g2, db2));
  }

  for (int off = 16; off > 0; off >>= 1) jacc += __shfl_xor(jacc, off, 32);
  if (lane == 0) Jl[wave] = jacc;  // J[a,b] complete within this block
  __syncthreads();

  if (tid < NBF) {
    float kc = 0.0f;
    for (int l = 0; l < 32; ++l) kc += Kacc[tid * 32 + l];
    atomicAdd(&Kg[a * NBF + tid], kc);  // combine 4 b-slices
  }
  if (tid < 24) Jg[a * NBF + slice * 24 + tid] = Jl[tid];
}

// --------------------------------------------------------------------------
// 96x96x96 f32 GEMM in LDS using v_wmma_f32_16x16x4_f32.
// 36 output tiles (6x6) across 16 waves; 24 WMMA per tile (K in steps of 4).
// Fragment layouts per ISA 7.12.2:
//   A 16x4:  lane = M (mod 16), K = 2*(lane/16) + vgpr
//   B 4x16:  lane = N (mod 16), K = 2*(lane/16) + vgpr  (mirror of A)
//   C/D:     lane = N (mod 16), M = vgpr + 8*(lane/16)
// --------------------------------------------------------------------------
__device__ inline void gemm96(const float* Aw, const float* Bw, float* Cw,
                              int tid, bool transB) {
  const int wave = tid >> 5, lane = tid & 31;
  const int half = lane >> 4, r = lane & 15;
  for (int tile = wave; tile < 36; tile += 16) {
    const int ti = tile / 6, tj = tile % 6;
    v8f acc = {0.f, 0.f, 0.f, 0.f, 0.f, 0.f, 0.f, 0.f};
    for (int k0 = 0; k0 < NBF; k0 += 4) {
      const int ka = k0 + 2 * half;
      v2f af = {Aw[(ti * 16 + r) * NBF + ka], Aw[(ti * 16 + r) * NBF + ka + 1]};
      v2f bf;
      if (!transB) {
        bf.x = Bw[ka * NBF + tj * 16 + r];
        bf.y = Bw[(ka + 1) * NBF + tj * 16 + r];
      } else {  // B = M^T : B[k][n] = M[n][k]
        bf.x = Bw[(tj * 16 + r) * NBF + ka];
        bf.y = Bw[(tj * 16 + r) * NBF + ka + 1];
      }
      acc = __builtin_amdgcn_wmma_f32_16x16x4_f32(
          false, af, false, bf, (short)0, acc, false, false);
    }
    for (int v = 0; v < 8; ++v)
      Cw[(ti * 16 + v + 8 * half) * NBF + tj * 16 + r] = acc[v];
  }
}

// --------------------------------------------------------------------------
// SCF kernel (single workgroup, 512 threads, ~149 KB LDS):
//   F = H + 2J - K ; Fp = A F A (WMMA) ; Jacobi eigh(Fp) ;
//   P = V24 V24^T ; D = A P A (WMMA) ; E = sum((F+H).*D) + Enuc
// --------------------------------------------------------------------------
__global__ __launch_bounds__(512) void scf_kernel(
    const float* __restrict__ H, const float* __restrict__ A,
    const float* __restrict__ Jg, const float* __restrict__ Kg,
    const float* __restrict__ Enuc, float* __restrict__ Dout,
    float* __restrict__ Eout) {
  __shared__ float As[NBF2], Xs[NBF2], Ys[NBF2], Zs[NBF2];  // 4 x 36 KB
  __shared__ float c_arr[48], s_arr[48];
  __shared__ int p_arr[48], q_arr[48];
  __shared__ float ev[NBF];
  __shared__ int selv[NBF];
  __shared__ float wred[16];
  const int tid = threadIdx.x;

  // Load A, build F = H + 2J - K into Xs
  for (int i = tid; i < NBF2; i += 512) {
    As[i] = A[i];
    Xs[i] = H[i] + 2.0f * Jg[i] - Kg[i];
  }
  __syncthreads();
  gemm96(As, Xs, Ys, tid, false);  // T1 = A*F
  __syncthreads();
  gemm96(Ys, As, Zs, tid, false);  // Fp = T1*A
  __syncthreads();

  // V = I (in Xs; F no longer needed — recomputed for energy)
  for (int i = tid; i < NBF2; i += 512)
    Xs[i] = ((i / NBF) == (i % NBF)) ? 1.0f : 0.0f;
  __syncthreads();

  // Parallel cyclic Jacobi: 48 disjoint rotations per round (circle method)
  for (int sweep = 0; sweep < NSWEEPS; ++sweep) {
    for (int r = 0; r < 95; ++r) {
      if (tid < 48) {
        int p, q;
        if (tid == 0) { p = 95; q = r; }
        else { p = (r + tid) % 95; q = (r + 95 - tid) % 95; }
        p_arr[tid] = p; q_arr[tid] = q;
        float apq = Zs[p * NBF + q];
        float cc = 1.0f, ss = 0.0f;
        if (fabsf(apq) > 1e-20f) {
          float tau = (Zs[q * NBF + q] - Zs[p * NBF + p]) / (2.0f * apq);
          float t = ((tau >= 0.0f) ? 1.0f : -1.0f) /
                    (fabsf(tau) + sqrtf(1.0f + tau * tau));
          cc = rsqrtf(1.0f + t * t);
          ss = t * cc;
        }
        c_arr[tid] = cc; s_arr[tid] = ss;
      }
      __syncthreads();
      // Column update on Zs (A*R) and eigenvector accumulation V*R
      for (int id = tid; id < 48 * NBF; id += 512) {
        int k = id / NBF, i = id % NBF;
        int p = p_arr[k], q = q_arr[k];
        float cc = c_arr[k], ss = s_arr[k];
        float zp = Zs[i * NBF + p], zq = Zs[i * NBF + q];
        Zs[i * NBF + p] = cc * zp - ss * zq;
        Zs[i * NBF + q] = ss * zp + cc * zq;
        float vp = Xs[i * NBF + p], vq = Xs[i * NBF + q];
        Xs[i * NBF + p] = cc * vp - ss * vq;
        Xs[i * NBF + q] = ss * vp + cc * vq;
      }
      __syncthreads();
      // Row update on Zs (R^T * B)
      for (int id = tid; id < 48 * NBF; id += 512) {
        int k = id / NBF, j = id % NBF;
        int p = p_arr[k], q = q_arr[k];
        float cc = c_arr[k], ss = s_arr[k];
        float zp = Zs[p * NBF + j], zq = Zs[q * NBF + j];
        Zs[p * NBF + j] = cc * zp - ss * zq;
        Zs[q * NBF + j] = ss * zp + cc * zq;
      }
      __syncthreads();
    }
  }

  // Rank-select the NDOCC lowest eigenvalues (O(N^2) comparisons)
  if (tid < NBF) ev[tid] = Zs[tid * NBF + tid];
  __syncthreads();
  if (tid < NBF) {
    int rank = 0;
    float lj = ev[tid];
    for (int i = 0; i < NBF; ++i) {
      float li = ev[i];
      if (li < lj || (li == lj && i < tid)) rank++;
    }
    selv[tid] = (rank < NDOCC) ? 1 : 0;
  }
  __syncthreads();
  // Zero non-occupied columns of V
  for (int i = tid; i < NBF2; i += 512)
    if (!selv[i % NBF]) Xs[i] = 0.0f;
  __syncthreads();

  gemm96(Xs, Xs, Ys, tid, true);   // P  = Vocc * Vocc^T
  __syncthreads();
  gemm96(As, Ys, Zs, tid, false);  // T2 = A*P
  __syncthreads();
  gemm96(Zs, As, Ys, tid, false);  // Dn = T2*A
  __syncthreads();

  // Store D_new, compute E = sum((F+H).*Dn) + Enuc  (F+H = 2H + 2J - K)
  float e = 0.0f;
  for (int i = tid; i < NBF2; i += 512) {
    float dn = Ys[i];
    Dout[i] = dn;
    e += (2.0f * H[i] + 2.0f * Jg[i] - Kg[i]) * dn;
  }
  for (int off = 16; off > 0; off >>= 1) e += __shfl_xor(e, off, 32);
  const int wave = tid >> 5, lane = tid & 31;
  if (lane == 0) wred[wave] = e;
  __syncthreads();
  if (tid == 0) {
    float tot = 0.0f;
    for (int w = 0; w < 16; ++w) tot += wred[w];
    Eout[0] = tot + Enuc[0];
  }
}

// --------------------------------------------------------------------------
extern "C" void kernel_launch(void* const* d_in, const int* in_sizes, int n_in,
                              void* d_out, int out_size, void* d_ws,
                              size_t ws_size, hipStream_t stream) {
  const float* H = (const float*)d_in[0];
  const float* A = (const float*)d_in[1];
  const float* G = (const float*)d_in[2];
  const float* Enuc = (const float*)d_in[3];
  // d_in[4] = ndocc (== 24, hardcoded)

  float* ws = (float*)d_ws;
  float* D  = ws;             // 9216 floats
  float* Jg = ws + NBF2;      // 9216 floats
  float* Kg = ws + 2 * NBF2;  // 9216 floats

  hipMemsetAsync(D, 0, NBF2 * sizeof(float), stream);  // D0 = 0
  for (int it = 0; it < N_ITERS; ++it) {
    hipMemsetAsync(Kg, 0, NBF2 * sizeof(float), stream);
    jk_kernel<<<dim3(96, 4), 768, 0, stream>>>(G, D, Jg, Kg);
    scf_kernel<<<1, 512, 0, stream>>>(H, A, Jg, Kg, Enuc, D, (float*)d_out);
  }
}